// ReDoTransformer_37074157699408
// MI455X (gfx1250) — compile-verified
//
#include <hip/hip_runtime.h>
#include <math.h>

// ---------------- problem constants (from reference) ----------------
#define BATCH   2
#define SEQ     4096
#define DMODEL  2048
#define NH      16
#define DK      128
#define DV      128
#define DHID    8192
#define SEGLEN  512
#define NSEG    8
#define MROWS   (BATCH*SEQ)        // 8192

typedef __attribute__((ext_vector_type(2))) float v2f;
typedef __attribute__((ext_vector_type(8))) float v8f;

// =====================================================================
// Tiled fp32 WMMA GEMM: C[M,N] = A[M,K] @ B[K,N] + bias, fused epilogues
//   mode 0: C = val
//   mode 1: C = val, C2 = elu(val)+1        (Q/K projections -> sigma)
//   mode 2: C = gelu(val)                   (FFN1, exact erf gelu)
//   mode 3: C = val + Res[row,col]          (FFN2 + residual x)
// 256 threads = 8 waves; each wave owns a 32x64 region = 2x4 subtiles
// of 16x16, accumulated with V_WMMA_F32_16X16X4_F32.
// A tile is staged with GLOBAL_LOAD_ASYNC_TO_LDS_B128 (ASYNCcnt DMA);
// B tile is staged transposed so both WMMA fragments are contiguous
// 8-byte LDS reads (no register repacking).
// =====================================================================
__global__ __launch_bounds__(256)
void gemm_wmma_f32(const float* __restrict__ A, const float* __restrict__ Bw,
                   const float* __restrict__ bias, float* __restrict__ C,
                   float* __restrict__ C2, const float* __restrict__ Res,
                   int M, int N, int K, int mode)
{
    // A rows padded to 36 floats (144B = 16B aligned for async B128 chunks)
    __shared__ float As[128][36];
    // B stored transposed: BsT[col][k], +1 pad keeps frag reads conflict-free
    __shared__ float BsT[128][33];

    const int t    = threadIdx.x;
    const int lane = t & 31;
    const int half = lane >> 4;     // K-pair select (A/B 16x4 layout)
    const int l16  = lane & 15;
    const int wid  = t >> 5;
    const int wm   = wid >> 1;      // 0..3 : wave row (32 rows each)
    const int wn   = wid & 1;       // 0..1 : wave col (64 cols each)
    const int tile_n = blockIdx.x * 128;
    const int tile_m = blockIdx.y * 128;

    const unsigned long long abase = (unsigned long long)A;

    v8f acc[2][4] = {};             // 2x4 16x16 f32 subtiles

    for (int kb = 0; kb < K; kb += 32) {
        // ---- stage A tile [128 x 32] via async DMA to LDS ----
        #pragma unroll
        for (int i = 0; i < 4; i++) {
            int j = t + 256 * i;                 // 1024 float4 chunks
            int row = j >> 3, c4 = (j & 7) << 2;
            unsigned lds = (unsigned)(unsigned long long)(uintptr_t)&As[row][c4];
            int goff = (int)((((size_t)(tile_m + row) * (size_t)K + (size_t)(kb + c4)) * 4u));
            asm volatile("global_load_async_to_lds_b128 %0, %1, %2"
                         :: "v"(lds), "v"(goff), "s"(abase) : "memory");
        }
        // ---- stage B tile [32 x 128], transposed into BsT ----
        #pragma unroll
        for (int i = 0; i < 4; i++) {
            int j = t + 256 * i;
            int row = j >> 5, c4 = (j & 31) << 2;
            const float* src = Bw + (size_t)(kb + row) * N + tile_n + c4;
            float b0 = src[0], b1 = src[1], b2 = src[2], b3 = src[3];
            BsT[c4 + 0][row] = b0;
            BsT[c4 + 1][row] = b1;
            BsT[c4 + 2][row] = b2;
            BsT[c4 + 3][row] = b3;
        }
        // wait for async A-tile DMA, then barrier for the whole tile
        asm volatile("s_wait_asynccnt 0x0" ::: "memory");
        __syncthreads();

        // prefetch next B k-tile into cache (global_prefetch_b8)
        if (kb + 32 < K) {
            __builtin_prefetch(Bw + (size_t)(kb + 32 + (t >> 3)) * N + tile_n + ((t & 7) << 4), 0, 1);
        }

        // ---- 8 x (2x4) WMMA_F32_16X16X4_F32 ----
        #pragma unroll
        for (int kk = 0; kk < 32; kk += 4) {
            v2f afr[2], bfr[4];
            #pragma unroll
            for (int i = 0; i < 2; i++) {
                int row = wm * 32 + i * 16 + l16;
                afr[i].x = As[row][kk + 2 * half];
                afr[i].y = As[row][kk + 2 * half + 1];
            }
            #pragma unroll
            for (int jn = 0; jn < 4; jn++) {
                int col = wn * 64 + jn * 16 + l16;
                bfr[jn].x = BsT[col][kk + 2 * half];
                bfr[jn].y = BsT[col][kk + 2 * half + 1];
            }
            #pragma unroll
            for (int i = 0; i < 2; i++)
                #pragma unroll
                for (int jn = 0; jn < 4; jn++)
                    acc[i][jn] = __builtin_amdgcn_wmma_f32_16x16x4_f32(
                        false, afr[i], false, bfr[jn],
                        (short)0, acc[i][jn], false, false);
        }
        __syncthreads();
    }

    // ---- epilogue (C layout: vgpr r -> M = r + 8*half, N = l16) ----
    #pragma unroll
    for (int i = 0; i < 2; i++) {
        #pragma unroll
        for (int jn = 0; jn < 4; jn++) {
            int base_m = tile_m + wm * 32 + i * 16 + 8 * half;
            int base_n = tile_n + wn * 64 + jn * 16 + l16;
            float bcol = bias[base_n];
            #pragma unroll
            for (int r = 0; r < 8; r++) {
                size_t idx = (size_t)(base_m + r) * N + base_n;
                float val = acc[i][jn][r] + bcol;
                if (mode == 1) {
                    C[idx]  = val;
                    C2[idx] = (val > 0.f) ? (val + 1.f) : __expf(val);  // elu+1
                } else if (mode == 2) {
                    C[idx] = 0.5f * val * (1.f + erff(val * 0.70710678118f));
                } else if (mode == 3) {
                    C[idx] = val + Res[idx];
                } else {
                    C[idx] = val;
                }
            }
        }
    }
}

// =====================================================================
// init compressive-memory state: mem = 0, z = 1/DK  (every launch)
// =====================================================================
__global__ void init_state(float* __restrict__ mem, float* __restrict__ z)
{
    int i = blockIdx.x * 256 + threadIdx.x;
    if (i < BATCH * NH * DK * DV) mem[i] = 0.f;
    if (i < BATCH * NH * DK)      z[i] = 1.0f / (float)DK;
}

// =====================================================================
// attn_local: one block = (b,h, 32-row q block) within a segment.
// Flash-style online softmax over causal k/v chunks of 32 rows, plus
// compressive-memory retrieval sigma(q)@mem / sigma(q)@z, gated mix.
// =====================================================================
__global__ __launch_bounds__(256)
void attn_local(const float* __restrict__ q,  const float* __restrict__ sq,
                const float* __restrict__ k,  const float* __restrict__ v,
                const float* __restrict__ mem,const float* __restrict__ z,
                const float* __restrict__ betas, float* __restrict__ att,
                int seg)
{
    __shared__ float qs[32][128];
    __shared__ float ks[32][128];
    __shared__ float vs[32][128];
    __shared__ float sbuf[32][33];
    __shared__ float rowmax[32], rowsum[32], rescale[32];
    __shared__ float zs[128];
    __shared__ float den[32];

    const int t  = threadIdx.x;
    const int rb = blockIdx.x & 15;           // 16 row-blocks of 32
    const int h  = (blockIdx.x >> 4) & 15;
    const int b  = blockIdx.x >> 8;
    const int l  = t >> 3;                    // row 0..31
    const int c0 = (t & 7) * 16;              // 16 DV cols per thread

    const int row0  = b * SEQ + seg * SEGLEN + rb * 32;  // q rows
    const int mrow0 = b * SEQ + seg * SEGLEN;            // k/v rows
    const int col0  = h * DK;

    // load q rows
    for (int idx = t; idx < 1024; idx += 256) {
        int r = idx >> 5, c = (idx & 31) << 2;
        const float* src = q + (size_t)(row0 + r) * (NH * DK) + col0 + c;
        qs[r][c] = src[0]; qs[r][c+1] = src[1]; qs[r][c+2] = src[2]; qs[r][c+3] = src[3];
    }
    if (t < 32) { rowmax[t] = -1e30f; rowsum[t] = 0.f; }
    float accv[16];
    #pragma unroll
    for (int i = 0; i < 16; i++) accv[i] = 0.f;
    __syncthreads();

    // ---------- phase A: causal local attention (online softmax) ----------
    for (int c = 0; c <= rb; c++) {           // causal: skip chunks > rb
        for (int idx = t; idx < 1024; idx += 256) {
            int r = idx >> 5, cc = (idx & 31) << 2;
            const float* sk_ = k + (size_t)(mrow0 + c*32 + r) * (NH*DK) + col0 + cc;
            ks[r][cc] = sk_[0]; ks[r][cc+1] = sk_[1]; ks[r][cc+2] = sk_[2]; ks[r][cc+3] = sk_[3];
            const float* sv_ = v + (size_t)(mrow0 + c*32 + r) * (NH*DV) + col0 + cc;
            vs[r][cc] = sv_[0]; vs[r][cc+1] = sv_[1]; vs[r][cc+2] = sv_[2]; vs[r][cc+3] = sv_[3];
        }
        __syncthreads();

        for (int e = t; e < 1024; e += 256) {  // scores 32x32
            int sl = e >> 5, sm = e & 31;
            float s;
            if (c * 32 + sm > rb * 32 + sl) {
                s = -1e9f;                      // causal mask
            } else {
                float a = 0.f;
                for (int kk = 0; kk < 128; kk++) a += qs[sl][kk] * ks[sm][kk];
                s = a * 0.08838834764831845f;   // 1/sqrt(128)
            }
            sbuf[sl][sm] = s;
        }
        __syncthreads();

        if (t < 32) {                           // per-row online softmax stats
            float omx = rowmax[t], cmx = -1e30f;
            for (int m = 0; m < 32; m++) cmx = fmaxf(cmx, sbuf[t][m]);
            float nmx = fmaxf(omx, cmx);
            float rs  = __expf(omx - nmx);
            float s   = 0.f;
            for (int m = 0; m < 32; m++) { float e2 = __expf(sbuf[t][m] - nmx); sbuf[t][m] = e2; s += e2; }
            rowsum[t]  = rowsum[t] * rs + s;
            rowmax[t]  = nmx;
            rescale[t] = rs;
        }
        __syncthreads();

        {
            float rs = rescale[l];
            #pragma unroll
            for (int i = 0; i < 16; i++) {
                float a = accv[i] * rs;
                for (int m = 0; m < 32; m++) a += sbuf[l][m] * vs[m][c0 + i];
                accv[i] = a;
            }
        }
        __syncthreads();
    }

    // ---------- phase B: memory retrieval ----------
    for (int idx = t; idx < 1024; idx += 256) {   // qs <- sigma(q)
        int r = idx >> 5, c = (idx & 31) << 2;
        const float* src = sq + (size_t)(row0 + r) * (NH * DK) + col0 + c;
        qs[r][c] = src[0]; qs[r][c+1] = src[1]; qs[r][c+2] = src[2]; qs[r][c+3] = src[3];
    }
    if (t < 128) zs[t] = z[(b * NH + h) * DK + t];
    __syncthreads();

    if (t < 32) {
        float d = 0.f;
        for (int kk = 0; kk < 128; kk++) d += qs[t][kk] * zs[kk];
        den[t] = d;
    }
    float num[16];
    #pragma unroll
    for (int i = 0; i < 16; i++) num[i] = 0.f;

    const float* memb = mem + (size_t)(b * NH + h) * DK * DV;
    for (int mc = 0; mc < 4; mc++) {
        for (int idx = t; idx < 1024; idx += 256) {
            int r = idx >> 5, c = (idx & 31) << 2;
            const float* src = memb + (size_t)(mc * 32 + r) * DV + c;
            ks[r][c] = src[0]; ks[r][c+1] = src[1]; ks[r][c+2] = src[2]; ks[r][c+3] = src[3];
        }
        __syncthreads();
        #pragma unroll
        for (int i = 0; i < 16; i++) {
            float a = num[i];
            for (int kk = 0; kk < 32; kk++) a += qs[l][mc * 32 + kk] * ks[kk][c0 + i];
            num[i] = a;
        }
        __syncthreads();
    }

    const float dl = den[l];
    const float rsum = rowsum[l];
    #pragma unroll
    for (int i = 0; i < 16; i++) {
        int vc = c0 + i;
        float g  = 1.f / (1.f + __expf(-betas[h * DV + vc]));   // sigmoid gate
        float am = num[i] / dl;
        float ad = accv[i] / rsum;
        att[(size_t)(row0 + l) * (NH * DV) + col0 + vc] = g * am + (1.f - g) * ad;
    }
}

// =====================================================================
// mem_update: delta rule for one (b,h):
//   v_del = v - sigma(k)@mem0 / sigma(k)@z0
//   mem  += sigma(k)^T @ v_del ;  z += sum_l sigma(k)
// Delta-mem held in 64 regs/thread; mem0 read through L2 (2MB total).
// =====================================================================
__global__ __launch_bounds__(256)
void mem_update(const float* __restrict__ sk, const float* __restrict__ v,
                float* __restrict__ mem, float* __restrict__ z, int seg)
{
    __shared__ float sks[32][128];
    __shared__ float vds[32][128];
    __shared__ float zs[128];
    __shared__ float den[32];

    const int t  = threadIdx.x;
    const int h  = blockIdx.x & 15;
    const int b  = blockIdx.x >> 4;
    const int kr = t >> 1;                 // mem row owned by thread
    const int vh = t & 1;
    const int c0 = vh * 64;

    float dmem[64];
    #pragma unroll
    for (int i = 0; i < 64; i++) dmem[i] = 0.f;
    float dz = 0.f;

    const size_t mbase = (size_t)(b * NH + h) * DK * DV;
    const int    zbase = (b * NH + h) * DK;
    if (t < 128) zs[t] = z[zbase + t];
    __syncthreads();

    const int row0 = b * SEQ + seg * SEGLEN;
    for (int lc = 0; lc < 16; lc++) {
        for (int idx = t; idx < 1024; idx += 256) {
            int r = idx >> 5, c = (idx & 31) << 2;
            const float* s1 = sk + (size_t)(row0 + lc*32 + r) * (NH*DK) + h*DK + c;
            sks[r][c] = s1[0]; sks[r][c+1] = s1[1]; sks[r][c+2] = s1[2]; sks[r][c+3] = s1[3];
            const float* s2 = v  + (size_t)(row0 + lc*32 + r) * (NH*DV) + h*DV + c;
            vds[r][c] = s2[0]; vds[r][c+1] = s2[1]; vds[r][c+2] = s2[2]; vds[r][c+3] = s2[3];
        }
        __syncthreads();

        if (t < 32) {
            float d = 0.f;
            for (int kk = 0; kk < 128; kk++) d += sks[t][kk] * zs[kk];
            den[t] = d;
        }
        __syncthreads();

        for (int e = t; e < 4096; e += 256) {     // v_del in place
            int l = e >> 7, vc = e & 127;
            float r2 = 0.f;
            for (int kk = 0; kk < 128; kk++)
                r2 += sks[l][kk] * mem[mbase + (size_t)kk * DV + vc];
            vds[l][vc] = vds[l][vc] - r2 / den[l];
        }
        __syncthreads();

        for (int l = 0; l < 32; l++) {            // rank-32 outer-product
            float s = sks[l][kr];
            if (vh == 0) dz += s;
            #pragma unroll
            for (int c = 0; c < 64; c++) dmem[c] += s * vds[l][c0 + c];
        }
        __syncthreads();
    }

    #pragma unroll
    for (int c = 0; c < 64; c++)
        mem[mbase + (size_t)kr * DV + c0 + c] += dmem[c];
    if (vh == 0) z[zbase + kr] += dz;
}

// =====================================================================
// Row LayerNorm: out = (y - mu) * rsqrt(var+1e-5) * w + b
// =====================================================================
__global__ __launch_bounds__(256)
void layernorm(const float* __restrict__ y, const float* __restrict__ w,
               const float* __restrict__ bias, float* __restrict__ out)
{
    __shared__ float red[256], red2[256];
    const int row = blockIdx.x;
    const float* yr = y + (size_t)row * DMODEL;
    float vals[8], s = 0.f, s2 = 0.f;
    #pragma unroll
    for (int i = 0; i < 8; i++) {
        float vv = yr[threadIdx.x + 256 * i];
        vals[i] = vv; s += vv; s2 += vv * vv;
    }
    red[threadIdx.x] = s; red2[threadIdx.x] = s2;
    __syncthreads();
    for (int off = 128; off > 0; off >>= 1) {
        if (threadIdx.x < off) {
            red[threadIdx.x]  += red[threadIdx.x + off];
            red2[threadIdx.x] += red2[threadIdx.x + off];
        }
        __syncthreads();
    }
    float mu  = red[0] / (float)DMODEL;
    float var = red2[0] / (float)DMODEL - mu * mu;
    float inv = rsqrtf(var + 1e-5f);
    #pragma unroll
    for (int i = 0; i < 8; i++) {
        int c = threadIdx.x + 256 * i;
        out[(size_t)row * DMODEL + c] = (vals[i] - mu) * inv * w[c] + bias[c];
    }
}

// =====================================================================
extern "C" void kernel_launch(void* const* d_in, const int* in_sizes, int n_in,
                              void* d_out, int out_size, void* d_ws, size_t ws_size,
                              hipStream_t stream)
{
    const float* x     = (const float*)d_in[0];
    const float* Wq    = (const float*)d_in[1];
    const float* bq    = (const float*)d_in[2];
    const float* Wk    = (const float*)d_in[3];
    const float* bk    = (const float*)d_in[4];
    const float* Wv    = (const float*)d_in[5];
    const float* bv    = (const float*)d_in[6];
    const float* Wo    = (const float*)d_in[7];
    const float* bo    = (const float*)d_in[8];
    const float* betas = (const float*)d_in[9];
    const float* W1    = (const float*)d_in[10];
    const float* b1    = (const float*)d_in[11];
    const float* W2    = (const float*)d_in[12];
    const float* b2    = (const float*)d_in[13];
    const float* ln_w  = (const float*)d_in[14];
    const float* ln_b  = (const float*)d_in[15];

    float* ws = (float*)d_ws;
    const size_t MD = (size_t)MROWS * DMODEL;      // 16M floats
    float* qb   = ws;
    float* sqb  = ws + 1 * MD;
    float* kb   = ws + 2 * MD;
    float* skb  = ws + 3 * MD;
    float* vb   = ws + 4 * MD;
    float* attb = ws + 5 * MD;
    float* hb   = ws + 6 * MD;                     // 8192 x 8192
    float* memb = hb + (size_t)MROWS * DHID;
    float* zb   = memb + (size_t)BATCH * NH * DK * DV;
    float* ab   = qb;                              // alias: q dead after attn
    float* yb   = kb;                              // alias: k dead after attn

    dim3 blk(256);
    dim3 gP(DMODEL / 128, MROWS / 128);            // 16 x 64

    // QKV projections (+ sigma for q,k fused)
    gemm_wmma_f32<<<gP, blk, 0, stream>>>(x, Wq, bq, qb, sqb, nullptr, MROWS, DMODEL, DMODEL, 1);
    gemm_wmma_f32<<<gP, blk, 0, stream>>>(x, Wk, bk, kb, skb, nullptr, MROWS, DMODEL, DMODEL, 1);
    gemm_wmma_f32<<<gP, blk, 0, stream>>>(x, Wv, bv, vb, nullptr, nullptr, MROWS, DMODEL, DMODEL, 0);

    init_state<<<2048, blk, 0, stream>>>(memb, zb);

    // sequential segments (stream order carries the mem/z recurrence)
    for (int s = 0; s < NSEG; s++) {
        attn_local<<<BATCH * NH * (SEGLEN / 32), blk, 0, stream>>>(
            qb, sqb, kb, vb, memb, zb, betas, attb, s);
        mem_update<<<BATCH * NH, blk, 0, stream>>>(skb, vb, memb, zb, s);
    }

    // output projection
    gemm_wmma_f32<<<gP, blk, 0, stream>>>(attb, Wo, bo, ab, nullptr, nullptr, MROWS, DMODEL, DMODEL, 0);

    // FFN1 with exact GELU
    dim3 g1(DHID / 128, MROWS / 128);              // 64 x 64
    gemm_wmma_f32<<<g1, blk, 0, stream>>>(ab, W1, b1, hb, nullptr, nullptr, MROWS, DHID, DMODEL, 2);

    // FFN2 + residual x
    dim3 g2(DMODEL / 128, MROWS / 128);
    gemm_wmma_f32<<<g2, blk, 0, stream>>>(hb, W2, b2, yb, nullptr, x, MROWS, DMODEL, DHID, 3);

    // LayerNorm -> output
    layernorm<<<MROWS, blk, 0, stream>>>(yb, ln_w, ln_b, (float*)d_out);
}